// Top_Left_Corner_66623532695931
// MI455X (gfx1250) — compile-verified
//
#include <hip/hip_runtime.h>

// CornerNet top+left corner pooling, fp32, planes of 128x128.
// out[h][w] = (max_{h'>=h} x[h'][w]) + (max_{w'>=w} x[h][w'])
//
// Memory-bound: 268MB in + 268MB out => ~23us floor at 23.3 TB/s.
// Strategy per block (one 64KB plane, 128 threads / 4 waves):
//   A) async-copy plane -> LDS (global_load_async_to_lds_b128, ASYNCcnt)
//   B) in-place reverse max-scan along W in LDS (thread = row)
//   C) reverse max-scan along H streaming rows from global (L2-resident
//      re-read, coalesced, NT hint = last use), fuse add + NT coalesced store.

#define H_DIM 128
#define W_DIM 128
#define SX    132          // padded LDS row stride in floats (16B-aligned rows)
#define NT_THREADS 128     // threads per block (4 waves, wave32)

#if defined(__has_builtin)
#  if __has_builtin(__builtin_amdgcn_global_load_async_to_lds_b128)
#    define USE_ASYNC_LDS 1
#  endif
#  if __has_builtin(__builtin_amdgcn_s_wait_asynccnt)
#    define USE_WAIT_ASYNC_BUILTIN 1
#  endif
#endif

typedef int v4i __attribute__((ext_vector_type(4)));

__device__ __forceinline__ void wait_asynccnt_zero() {
#if defined(USE_WAIT_ASYNC_BUILTIN)
  __builtin_amdgcn_s_wait_asynccnt(0);
#else
  asm volatile("s_wait_asynccnt 0" ::: "memory");
#endif
}

__global__ __launch_bounds__(NT_THREADS)
void corner_pool_kernel(const float* __restrict__ x, float* __restrict__ out) {
  __shared__ float tile[H_DIM * SX];   // 128*132*4 = 67,584 B

  const int tid = threadIdx.x;
  const long long plane = blockIdx.x;
  const float* __restrict__ gx = x + plane * (long long)(H_DIM * W_DIM);
  float* __restrict__ gout = out + plane * (long long)(H_DIM * W_DIM);

  // ---------------- Phase A: stage plane into LDS -------------------------
  // Lane layout: 32 lanes * float4 cover one 128-float row; 4 rows per pass.
  {
    const int w0 = (tid & 31) * 4;  // column start, multiple of 4
    const int h0 = tid >> 5;        // 0..3
#if defined(USE_ASYNC_LDS)
#pragma unroll
    for (int it = 0; it < 32; ++it) {
      const int h = h0 + 4 * it;
      __builtin_amdgcn_global_load_async_to_lds_b128(
          (v4i*)(gx + (long long)h * W_DIM + w0),
          (v4i*)(&tile[h * SX + w0]),
          0, 0);
    }
    wait_asynccnt_zero();
#else
#pragma unroll 8
    for (int it = 0; it < 32; ++it) {
      const int h = h0 + 4 * it;
      const float4 v = *(const float4*)(gx + (long long)h * W_DIM + w0);
      *(float4*)(&tile[h * SX + w0]) = v;
    }
#endif
  }
  __syncthreads();

  // ------------- Phase B: left pool, in-place reverse scan along W --------
  // thread t owns row t; read-then-overwrite the same slot (scan-safe).
  {
    const int h = tid;
    float m = -__builtin_inff();
#pragma unroll 8
    for (int w = W_DIM - 1; w >= 0; --w) {
      m = fmaxf(m, tile[h * SX + w]);
      tile[h * SX + w] = m;   // tile now holds left-pool result
    }
  }
  __syncthreads();

  // ------------- Phase C: top pool from global (L2 hit) + fused store -----
  // thread t owns column t; per step, lanes read/write one contiguous row.
  // Re-read of x is its LAST use -> NT load; output is write-once -> NT store.
  {
    const int w = tid;
    float m = -__builtin_inff();
#pragma unroll 8
    for (int h = H_DIM - 1; h >= 0; --h) {
      const float v = __builtin_nontemporal_load(gx + (long long)h * W_DIM + w);
      m = fmaxf(m, v);
      __builtin_nontemporal_store(m + tile[h * SX + w],
                                  gout + (long long)h * W_DIM + w);
    }
  }
}

extern "C" void kernel_launch(void* const* d_in, const int* in_sizes, int n_in,
                              void* d_out, int out_size, void* d_ws, size_t ws_size,
                              hipStream_t stream) {
  (void)n_in; (void)out_size; (void)d_ws; (void)ws_size;
  const float* x = (const float*)d_in[0];
  float* out = (float*)d_out;
  const int planes = in_sizes[0] / (H_DIM * W_DIM);   // 16*256 = 4096
  corner_pool_kernel<<<dim3(planes), dim3(NT_THREADS), 0, stream>>>(x, out);
}